// NsoltBlockIdct3dLayer_25074019074425
// MI455X (gfx1250) — compile-verified
//
#include <hip/hip_runtime.h>
#include <hip/hip_bf16.h>

// 3D block IDCT (4x4x4, orthonormal) recast as Out[1M,64] = In[1M,64] * W^T
// with the coefficient permutation folded into the constant 64x64 matrix W.
// fp32 WMMA (v_wmma_f32_16x16x4_f32) keeps exact fp32 semantics; the kernel
// is bandwidth-bound (512 MB traffic, ~22us floor at 23.3 TB/s).
//
// CDNA5-specific paths used:
//   - v_wmma_f32_16x16x4_f32            (matrix core, fp32)
//   - global_load_async_to_lds_b128     (ASYNCcnt DMA, global -> LDS, no VGPRs)
//   - s_wait_asynccnt                   (split async counter)
//   - non-temporal b128 stores          (touch-once stream, keep L2 clean)

typedef __attribute__((ext_vector_type(2))) float v2f;
typedef __attribute__((ext_vector_type(4))) float v4f;
typedef __attribute__((ext_vector_type(8))) float v8f;

// Orthonormal 4-point IDCT (DCT-III): ID[n][k] = s_k * cos(pi*(n+0.5)*k/4)
__device__ __constant__ float c_idct4[16] = {
    0.5f,  0.65328148243818826f,  0.5f,  0.27059805007309851f,
    0.5f,  0.27059805007309851f, -0.5f, -0.65328148243818826f,
    0.5f, -0.27059805007309851f, -0.5f,  0.65328148243818826f,
    0.5f, -0.65328148243818826f,  0.5f, -0.27059805007309851f,
};

__global__ __launch_bounds__(256)
void nsolt_idct3d_wmma(const float* __restrict__ in, float* __restrict__ out,
                       int nstrips) {
    // W[o][t], row-major, 64x64 fp32 = 16 KB
    __shared__ __align__(16) float sW[64 * 64];
    // per-wave staging: 16 blocks x 64 floats = 4 KB each (input, then output)
    __shared__ __align__(16) float sA[8][16 * 64];

    const int tid  = threadIdx.x;
    const int lane = tid & 31;
    const int wave = tid >> 5;

    // ---- Build W in LDS: W[o][t] = ID[i][fy]*ID[j][fx]*ID[k][fz] ----
    // o = i*16 + j*4 + k ; t -> (fy,fx,fz) via the reference permutation:
    //   q = t>>3 gives parity bits (ey,ex,ez); r = t&7 gives (y',x',z');
    //   f = 2*prime + parity
    for (int e = tid; e < 4096; e += 256) {
        const int o = e >> 6, c = e & 63;
        const int i = (o >> 4) & 3, j = (o >> 2) & 3, k = o & 3;
        const int q = c >> 3, r = c & 7;
        const int fy = (((r >> 2) & 1) << 1) | ((q >> 2) & 1);
        const int fx = (((r >> 1) & 1) << 1) | ((q >> 1) & 1);
        const int fz = ((r & 1) << 1) | (q & 1);
        sW[e] = c_idct4[i * 4 + fy] * c_idct4[j * 4 + fx] * c_idct4[k * 4 + fz];
    }
    __syncthreads();

    // wave-uniform strip id (keeps the base address in SGPRs for the async DMA)
    const int strip = __builtin_amdgcn_readfirstlane(blockIdx.x * 8 + wave);
    if (strip >= nstrips) return;   // whole-wave exit: EXEC stays all-1s

    const float* __restrict__ src = in  + (size_t)strip * 1024;
    float*       __restrict__ dst = out + (size_t)strip * 1024;
    float* aw = sA[wave];

    // ---- Async DMA: 16x64 contiguous fp32 tile, global -> LDS ----
    // VDST = per-lane LDS byte offset (low 32 bits of the generic shared
    // address ARE the DS offset: generic LDS addr = {SHARED_BASE, offset}).
    // GVS mode: mem = SADDR(SGPR pair) + VADDR(32-bit) + inst_offset,
    // same inst_offset also advances the LDS address.
    {
        const unsigned ldsAddr = (unsigned)(uintptr_t)(aw + lane * 4);
        const unsigned gOff    = lane * 16u;
        #pragma unroll
        for (int it = 0; it < 8; ++it) {
            asm volatile("global_load_async_to_lds_b128 %0, %1, %2 offset:%3"
                         :: "v"(ldsAddr), "v"(gOff), "s"(src), "i"(it * 512)
                         : "memory");
        }
    }
#if __has_builtin(__builtin_amdgcn_s_wait_asynccnt)
    __builtin_amdgcn_s_wait_asynccnt(0);
#else
    asm volatile("s_wait_asynccnt 0x0" ::: "memory");
#endif

    const int half = lane >> 4;   // 0: K = {0,1}; 1: K = {2,3}
    const int lm   = lane & 15;

    // ---- A fragments: 16x4 f32 tiles, lane lm holds row M=lm ----
    v2f afrag[16];
    #pragma unroll
    for (int kk = 0; kk < 16; ++kk)
        afrag[kk] = *(const v2f*)&aw[lm * 64 + kk * 4 + 2 * half];

    // ---- 4 output tiles of 16x16, K = 64 in 16 fp32-WMMA steps each ----
    #pragma unroll
    for (int t = 0; t < 4; ++t) {
        const int n0 = t * 16;
        v8f acc = {};
        #pragma unroll
        for (int kk = 0; kk < 16; ++kk) {
            // B fragment: B[k][n] = W[n0+lm][kk*4 + 2*half + {0,1}]
            v2f bfrag = *(const v2f*)&sW[(n0 + lm) * 64 + kk * 4 + 2 * half];
            acc = __builtin_amdgcn_wmma_f32_16x16x4_f32(
                false, afrag[kk], false, bfrag, (short)0, acc, false, false);
        }
        // D layout: VGPR v -> row m = v + 8*half, col N = n0 + lm.
        // Stage into LDS (input tile fully consumed; DS pipe is in-order
        // per wave, so no barrier needed for this wave-private region).
        #pragma unroll
        for (int v = 0; v < 8; ++v)
            aw[(v + 8 * half) * 64 + n0 + lm] = acc[v];
    }

    // ---- Contiguous non-temporal b128 stores: 4 KB per strip ----
    {
        const v4f* s4 = (const v4f*)aw;
        v4f*       d4 = (v4f*)dst;
        #pragma unroll
        for (int it = 0; it < 8; ++it) {
            v4f vv = s4[it * 32 + lane];
            __builtin_nontemporal_store(vv, d4 + it * 32 + lane);
        }
    }
}

extern "C" void kernel_launch(void* const* d_in, const int* in_sizes, int n_in,
                              void* d_out, int out_size, void* d_ws, size_t ws_size,
                              hipStream_t stream) {
    const float* x0 = (const float*)d_in[0];
    float* out = (float*)d_out;

    const long long n_elems = (long long)in_sizes[0];   // 4*64^3*64 = 67,108,864
    const long long nblocks = n_elems / 64;             // 1,048,576 blocks of 64
    const int nstrips = (int)((nblocks + 15) / 16);     // 16 blocks per wave
    const int grid    = (nstrips + 7) / 8;              // 8 waves per 256-thread WG

    nsolt_idct3d_wmma<<<grid, 256, 0, stream>>>(x0, out, nstrips);
}